// Conv_layer_60842506715659
// MI455X (gfx1250) — compile-verified
//
#include <hip/hip_runtime.h>
#include <math.h>

#define BS    8
#define V     2048
#define NNB   20
#define INC   64
#define OUTC  128
#define SUP   2
#define SUPC  (SUP * OUTC)          // 256
#define FEATC (OUTC + SUPC)         // 384
#define MROWS (BS * V)              // 16384
#define EPSF  1e-12f

typedef float v2f __attribute__((ext_vector_type(2)));
typedef float v8f __attribute__((ext_vector_type(8)));

// ---------------------------------------------------------------------------
// GEMM  C[M,N] = A[M,K] * B + bias[N]   using V_WMMA_F32_16X16X4_F32.
// BT=false: B is row-major [K,N].  BT=true: B given as [N,K] (i.e. use B^T).
// One wave computes one 16x16 tile of C.  blockDim = 128 (4 waves).
//
// Fragment layouts per CDNA5 ISA 7.12.2 (f32, wave32):
//   A 16x4 : lanes 0-15 -> M=lane,   v0=K0  v1=K1 ; lanes 16-31 -> v0=K2 v1=K3
//   B 4x16 : lanes 0-15 -> N=lane,   v0=K0  v1=K1 ; lanes 16-31 -> v0=K2 v1=K3
//   C 16x16: vgpr r -> M = r + 8*(lane>=16), N = lane&15
// ---------------------------------------------------------------------------
template <bool BT>
__global__ __launch_bounds__(128)
void gemm_bias_wmma(const float* __restrict__ A, const float* __restrict__ B,
                    const float* __restrict__ bias, float* __restrict__ C,
                    int M, int N, int K) {
  const int lane  = threadIdx.x & 31;
  const int wave  = threadIdx.x >> 5;
  const int tile  = blockIdx.x * 4 + wave;
  const int tilesN = N >> 4;
  const int tm = tile / tilesN;
  const int tn = tile - tm * tilesN;
  if (tm * 16 >= M) return;                 // wave-uniform: EXEC stays all-1s

  const int m0   = tm << 4;
  const int n0   = tn << 4;
  const int half = lane >> 4;               // 0 or 1
  const int l16  = lane & 15;

  v8f acc = {0.f, 0.f, 0.f, 0.f, 0.f, 0.f, 0.f, 0.f};

  const float* __restrict__ Arow = A + (size_t)(m0 + l16) * K;

  for (int k0 = 0; k0 < K; k0 += 4) {
    const int ka = k0 + 2 * half;
    v2f a;
    a.x = Arow[ka];
    a.y = Arow[ka + 1];

    v2f b;
    if (BT) {                               // B stored [N,K]; want B^T[k][n]
      const float* bp = B + (size_t)(n0 + l16) * K + ka;
      b.x = bp[0];
      b.y = bp[1];
    } else {                                // B stored [K,N]
      const float* bp = B + (size_t)ka * N + n0 + l16;
      b.x = bp[0];
      b.y = bp[N];
    }
    // (neg_a, A, neg_b, B, c_mod, C, reuse_a, reuse_b)
    acc = __builtin_amdgcn_wmma_f32_16x16x4_f32(false, a, false, b,
                                                (short)0, acc, false, false);
  }

  const float bv = bias[n0 + l16];
  float* __restrict__ Crow = C + (size_t)(m0 + 8 * half) * N + n0 + l16;
#pragma unroll
  for (int r = 0; r < 8; ++r)
    Crow[(size_t)r * N] = acc[r] + bv;
}

// ---------------------------------------------------------------------------
// Geometry + neighbor-gather + max/sup-sum stage.
// One block (256 threads) per vertex row; thread s owns support channel s.
// ---------------------------------------------------------------------------
__global__ __launch_bounds__(256)
void geom_fuse_kernel(const int* __restrict__ nbr, const float* __restrict__ verts,
                      const float* __restrict__ feat,        // [MROWS, 384]
                      const float* __restrict__ directions,  // [3, 256]
                      const float* __restrict__ distw,       // [256]
                      float* __restrict__ out256) {          // [MROWS, 256]
  const int row = blockIdx.x;                // b*V + v
  const int b   = row / V;
  const int s   = threadIdx.x;               // 0..255

  __shared__ float s_dirn[NNB][3];
  __shared__ float s_dist[NNB];
  __shared__ int   s_nidx[NNB];
  __shared__ float s_acc[SUPC];
  __shared__ float s_df[SUPC];

  if (s < NNB) {
    const int ni = nbr[(size_t)row * NNB + s];
    s_nidx[s] = ni;
    const float* pv = verts + (size_t)row * 3;
    const float* pn = verts + ((size_t)b * V + ni) * 3;
    const float dx = pn[0] - pv[0];
    const float dy = pn[1] - pv[1];
    const float dz = pn[2] - pv[2];
    const float d  = sqrtf(dx * dx + dy * dy + dz * dz);
    const float iv = 1.0f / fmaxf(d, EPSF);
    s_dirn[s][0] = dx * iv;
    s_dirn[s][1] = dy * iv;
    s_dirn[s][2] = dz * iv;
    s_dist[s] = d;
  }
  __syncthreads();

  float dmax = 0.0f;                         // dist >= 0
#pragma unroll
  for (int n = 0; n < NNB; ++n) dmax = fmaxf(dmax, s_dist[n]);

  // sdn column s = directions[:,s] / max(||directions[:,s]||, EPS)
  float d0 = directions[0 * SUPC + s];
  float d1 = directions[1 * SUPC + s];
  float d2 = directions[2 * SUPC + s];
  const float nrm = sqrtf(d0 * d0 + d1 * d1 + d2 * d2);
  const float iv  = 1.0f / fmaxf(nrm, EPSF);
  d0 *= iv; d1 *= iv; d2 *= iv;

  float acc = -INFINITY;
  const size_t bbase = (size_t)b * V;
#pragma unroll 4
  for (int n = 0; n < NNB; ++n) {
    const float th = fmaxf(0.0f, d0 * s_dirn[n][0] +
                                 d1 * s_dirn[n][1] +
                                 d2 * s_dirn[n][2]);
    const float sg = feat[(bbase + s_nidx[n]) * FEATC + OUTC + s];
    acc = fmaxf(acc, th * sg);
  }

  s_acc[s] = acc;
  s_df[s]  = fmaxf(0.0f, dmax * distw[s]);
  __syncthreads();

  if (s < OUTC) {
    const float center = feat[(size_t)row * FEATC + s];
    out256[(size_t)row * SUPC + s] = center + s_acc[s] + s_acc[s + OUTC];
  } else {
    const int c = s - OUTC;
    out256[(size_t)row * SUPC + OUTC + c] = s_df[c] + s_df[c + OUTC];
  }
}

// ---------------------------------------------------------------------------
extern "C" void kernel_launch(void* const* d_in, const int* in_sizes, int n_in,
                              void* d_out, int out_size, void* d_ws, size_t ws_size,
                              hipStream_t stream) {
  const int*   nbr   = (const int*)  d_in[0];   // (8,2048,20) int32
  const float* verts = (const float*)d_in[1];   // (8,2048,3)
  const float* fmap  = (const float*)d_in[2];   // (8,2048,64)
  const float* wgt   = (const float*)d_in[3];   // (64,384)
  const float* bias  = (const float*)d_in[4];   // (384)
  const float* dirs  = (const float*)d_in[5];   // (3,256)
  const float* distw = (const float*)d_in[6];   // (1,256)
  const float* mlpw  = (const float*)d_in[7];   // (128,256)
  const float* mlpb  = (const float*)d_in[8];   // (128)
  float* out = (float*)d_out;                   // (8,2048,128)

  float* feat   = (float*)d_ws;                         // MROWS x 384 (24 MB)
  float* out256 = feat + (size_t)MROWS * FEATC;         // MROWS x 256 (16 MB)

  // K1: feat = fmap @ weights + bias   (M=16384, K=64, N=384)
  {
    const int tiles = (MROWS / 16) * (FEATC / 16);      // 24576
    gemm_bias_wmma<false><<<tiles / 4, 128, 0, stream>>>(
        fmap, wgt, bias, feat, MROWS, FEATC, INC);
  }

  // K2: geometry + gather + max/sup-sum -> out256
  geom_fuse_kernel<<<MROWS, 256, 0, stream>>>(nbr, verts, feat, dirs, distw, out256);

  // K3: out = out256 @ mlp_w^T + mlp_b (M=16384, K=256, N=128)
  {
    const int tiles = (MROWS / 16) * (OUTC / 16);       // 8192
    gemm_bias_wmma<true><<<tiles / 4, 128, 0, stream>>>(
        out256, mlpw, mlpb, out, MROWS, OUTC, SUPC);
  }
}